// hier_softmax_73650099192241
// MI455X (gfx1250) — compile-verified
//
#include <hip/hip_runtime.h>
#include <hip/hip_bf16.h>

// Problem constants (match reference)
#define BRANCH   32
#define CDIM     64
#define BATCH    32768
#define NBUCK2   1024          // level-2 buckets = value >> 5
#define WROW     (BRANCH*CDIM) // 2048 floats per node weight tile

typedef __attribute__((ext_vector_type(2))) float v2f;
typedef __attribute__((ext_vector_type(8))) float v8f;

// ---------------------------------------------------------------------------
// Counting-sort helpers
// ---------------------------------------------------------------------------
__global__ void zero_i32_kernel(int* __restrict__ p, int n) {
    int i = blockIdx.x * blockDim.x + threadIdx.x;
    if (i < n) p[i] = 0;
}

__global__ void hist_kernel(const int* __restrict__ value, int* __restrict__ counts) {
    int e = blockIdx.x * blockDim.x + threadIdx.x;
    if (e < BATCH) atomicAdd(&counts[value[e] >> 5], 1);
}

__global__ void scan_kernel(const int* __restrict__ counts, int* __restrict__ start2) {
    if (threadIdx.x == 0 && blockIdx.x == 0) {
        int s = 0;
        for (int i = 0; i < NBUCK2; ++i) { start2[i] = s; s += counts[i]; }
        start2[NBUCK2] = s;
    }
}

__global__ void scatter_kernel(const int* __restrict__ value,
                               const int* __restrict__ start2,
                               int* __restrict__ cursor,
                               int* __restrict__ sortedIdx) {
    int e = blockIdx.x * blockDim.x + threadIdx.x;
    if (e < BATCH) {
        int g   = value[e] >> 5;
        int pos = atomicAdd(&cursor[g], 1);
        sortedIdx[start2[g] + pos] = e;
    }
}

// ---------------------------------------------------------------------------
// Grouped GEMM + softmax per level, fp32 WMMA (V_WMMA_F32_16X16X4_F32)
//   One block  = one node bucket  (W[node] staged once in LDS, 8 KB)
//   One wave   = one 16-element tile: logits[16x32] = ctx[16x64] @ W[node]^T
//   K=64 split into 16 WMMA K-steps of 4; two N-tiles of 16 -> 32 WMMAs/tile
// ---------------------------------------------------------------------------
template <int LEVEL>
__global__ __launch_bounds__(256)
void level_kernel(const float* __restrict__ ctx,
                  const int*   __restrict__ value,
                  const float* __restrict__ W,
                  const float* __restrict__ bias,
                  const int*   __restrict__ start2,
                  const int*   __restrict__ sortedIdx,
                  float*       __restrict__ probOut) {
    __shared__ float sW[WROW];          // staged W[node]: [BRANCH][CDIM], 8 KB
    __shared__ float slog[8][16 * 32];  // per-wave logits tile, 2 KB each

    const int bucket = blockIdx.x;
    int start, count, node;
    if (LEVEL == 0)      { start = 0;                  count = BATCH;                                    node = 0; }
    else if (LEVEL == 1) { start = start2[bucket*32];  count = start2[(bucket+1)*32] - start;            node = 1 + bucket; }
    else                 { start = start2[bucket];     count = start2[bucket+1] - start;                 node = 1 + BRANCH + bucket; }

    // Stage this node's weights into LDS (block-cooperative, coalesced)
    const float* wsrc = W + (long)node * WROW;
    for (int i = threadIdx.x; i < WROW; i += blockDim.x) sW[i] = wsrc[i];
    __syncthreads();

    if (count == 0) return;  // uniform per block

    const int wave = threadIdx.x >> 5;
    const int lane = threadIdx.x & 31;
    const int half = lane >> 4;   // which 16-lane half of the wave
    const int l16  = lane & 15;

    const int wavesTotal = (gridDim.y * blockDim.x) >> 5;
    const int waveGlobal = blockIdx.y * (blockDim.x >> 5) + wave;
    const int nTiles     = (count + 15) >> 4;

    float* lg = slog[wave];

    for (int tile = waveGlobal; tile < nTiles; tile += wavesTotal) {
        const int  rloc  = tile * 16 + l16;          // row within bucket
        const bool valid = rloc < count;
        const int  e     = sortedIdx[valid ? (start + rloc) : start]; // clamp pad rows
        const float* cp  = ctx + (long)e * CDIM;

        v8f c0 = {0.f,0.f,0.f,0.f,0.f,0.f,0.f,0.f};
        v8f c1 = c0;

        // K loop: 16 steps of K=4. A lane(l16=M, half picks K pair {0,1}/{2,3});
        // B lane(l16=N, same K striping) sourced from LDS.
        #pragma unroll 4
        for (int ks = 0; ks < CDIM / 4; ++ks) {
            const int ka = ks * 4 + half * 2;
            v2f a, b0, b1;
            a.x  = cp[ka];
            a.y  = cp[ka + 1];
            b0.x = sW[l16 * CDIM + ka];
            b0.y = sW[l16 * CDIM + ka + 1];
            b1.x = sW[(16 + l16) * CDIM + ka];
            b1.y = sW[(16 + l16) * CDIM + ka + 1];
            c0 = __builtin_amdgcn_wmma_f32_16x16x4_f32(false, a, false, b0,
                                                       (short)0, c0, false, false);
            c1 = __builtin_amdgcn_wmma_f32_16x16x4_f32(false, a, false, b1,
                                                       (short)0, c1, false, false);
        }

        // D layout: VGPR v -> row (v + 8*half), col l16 (tile0) / 16+l16 (tile1)
        #pragma unroll
        for (int v = 0; v < 8; ++v) {
            const int row = v + 8 * half;
            lg[row * 32 + l16]      = c0[v];
            lg[row * 32 + 16 + l16] = c1[v];
        }
        // Same-wave DS ops are in-order; reads below observe the writes above.

        if (lane < 16 && valid) {
            const float* brow = bias + node * BRANCH;
            float l[32];
            float mx = -3.402823e38f;
            #pragma unroll
            for (int j = 0; j < 32; ++j) {
                float t = lg[l16 * 32 + j] + brow[j];
                l[j] = t;
                mx = fmaxf(mx, t);
            }
            float sum = 0.f;
            #pragma unroll
            for (int j = 0; j < 32; ++j) sum += __expf(l[j] - mx);

            const int val = value[e];
            int digit;
            if (LEVEL == 0)      digit = val >> 10;
            else if (LEVEL == 1) digit = (val >> 5) & 31;
            else                 digit = val & 31;

            probOut[e] = __expf(l[digit] - mx) / sum;
        }
    }
}

__global__ void combine_kernel(const float* __restrict__ p0,
                               const float* __restrict__ p1,
                               const float* __restrict__ p2,
                               float* __restrict__ out) {
    int e = blockIdx.x * blockDim.x + threadIdx.x;
    if (e < BATCH) out[e] = p0[e] * p1[e] * p2[e];
}

// ---------------------------------------------------------------------------
extern "C" void kernel_launch(void* const* d_in, const int* in_sizes, int n_in,
                              void* d_out, int out_size, void* d_ws, size_t ws_size,
                              hipStream_t stream) {
    const float* ctx   = (const float*)d_in[0];   // [BATCH, CDIM]
    const int*   value = (const int*)  d_in[1];   // [BATCH]
    const float* W     = (const float*)d_in[2];   // [NNODES, BRANCH, CDIM]
    const float* bias  = (const float*)d_in[3];   // [NNODES, BRANCH]
    float*       out   = (float*)d_out;

    // Workspace layout
    char* ws = (char*)d_ws;
    int*   counts    = (int*)ws;                 ws += NBUCK2 * sizeof(int);
    int*   start2    = (int*)ws;                 ws += (NBUCK2 + 1) * sizeof(int);
    int*   cursor    = (int*)ws;                 ws += NBUCK2 * sizeof(int);
    int*   sortedIdx = (int*)ws;                 ws += BATCH * sizeof(int);
    float* p0        = (float*)ws;               ws += BATCH * sizeof(float);
    float* p1        = (float*)ws;               ws += BATCH * sizeof(float);
    float* p2        = (float*)ws;               ws += BATCH * sizeof(float);
    (void)ws_size; (void)in_sizes; (void)n_in; (void)out_size;

    // 1) counting sort by value>>5 (groups by node for both level 1 and 2)
    zero_i32_kernel<<<(2*NBUCK2 + 255)/256, 256, 0, stream>>>(counts, 2*NBUCK2); // counts+start2 head... zero counts & cursor
    // (counts and cursor are not adjacent; zero each explicitly)
    zero_i32_kernel<<<(NBUCK2 + 255)/256, 256, 0, stream>>>(cursor, NBUCK2);
    hist_kernel<<<(BATCH + 255)/256, 256, 0, stream>>>(value, counts);
    scan_kernel<<<1, 32, 0, stream>>>(counts, start2);
    scatter_kernel<<<(BATCH + 255)/256, 256, 0, stream>>>(value, start2, cursor, sortedIdx);

    // 2) per-level grouped GEMM (WMMA fp32) + softmax + digit select
    level_kernel<0><<<dim3(1, 64),   256, 0, stream>>>(ctx, value, W, bias, start2, sortedIdx, p0);
    level_kernel<1><<<dim3(32, 8),   256, 0, stream>>>(ctx, value, W, bias, start2, sortedIdx, p1);
    level_kernel<2><<<dim3(NBUCK2,1),256, 0, stream>>>(ctx, value, W, bias, start2, sortedIdx, p2);

    // 3) product along path
    combine_kernel<<<(BATCH + 255)/256, 256, 0, stream>>>(p0, p1, p2, out);
}